// NN_38010460570162
// MI455X (gfx1250) — compile-verified
//
#include <hip/hip_runtime.h>
#include <math.h>

typedef float v2f __attribute__((ext_vector_type(2)));
typedef float v8f __attribute__((ext_vector_type(8)));

#define B_DIM   256
#define IN_DIM  4096
#define OUT_DIM 4096
#define NBUCKET 256   // OUT_DIM / 16
#define PADK    16    // bucket capacity rounded to 16 edges (4 WMMAs), kills tail code

// ---------------- pass 1: zero bucket counters ----------------
__global__ void NN_zero_counts(int* __restrict__ counts) {
    int t = blockIdx.x * blockDim.x + threadIdx.x;
    if (t < NBUCKET) counts[t] = 0;
}

// ---------------- pass 2: histogram edges by out-row bucket ----------------
__global__ void NN_count_edges(const int* __restrict__ w_row,
                               int* __restrict__ counts, int nnz) {
    int e = blockIdx.x * blockDim.x + threadIdx.x;
    if (e < nnz) {
        int b = (w_row[e] >> 4) & (NBUCKET - 1);
        atomicAdd(&counts[b], 1);
    }
}

// ------- pass 3: exclusive scan over PADK-rounded capacities (1 block) ------
__global__ void NN_scan_offsets(const int* __restrict__ counts,
                                int* __restrict__ offsets,
                                int* __restrict__ cursor) {
    __shared__ int tmp[NBUCKET];
    int t = threadIdx.x;
    int cap = (counts[t] + (PADK - 1)) & ~(PADK - 1);
    tmp[t] = cap;
    __syncthreads();
    for (int d = 1; d < NBUCKET; d <<= 1) {
        int v = (t >= d) ? tmp[t - d] : 0;
        __syncthreads();
        tmp[t] += v;
        __syncthreads();
    }
    if (t == 0) offsets[0] = 0;
    offsets[t + 1] = tmp[t];          // padded inclusive -> offsets[b+1]
    cursor[t]      = tmp[t] - cap;    // padded exclusive -> scatter cursor
}

// -------- pass 4a: zero the pad records (val=0 => zero contribution) --------
__global__ void NN_fill_pad(const int* __restrict__ counts,
                            const int* __restrict__ offsets,
                            int2* __restrict__ rec) {
    int b = blockIdx.x * blockDim.x + threadIdx.x;
    if (b < NBUCKET) {
        int p = offsets[b] + counts[b];
        int e = offsets[b + 1];
        for (; p < e; ++p) rec[p] = make_int2(0, 0);
    }
}

// ---------------- pass 4b: scatter edges into fused 8B records --------------
// rec[pos] = { (row & 15) << 12 | col , bits(val) }
__global__ void NN_scatter_edges(const int* __restrict__ w_row,
                                 const int* __restrict__ w_col,
                                 const float* __restrict__ w_val,
                                 int* __restrict__ cursor,
                                 int2* __restrict__ rec, int nnz) {
    int e = blockIdx.x * blockDim.x + threadIdx.x;
    if (e < nnz) {
        int r = w_row[e];
        int b = (r >> 4) & (NBUCKET - 1);
        int pos = atomicAdd(&cursor[b], 1);
        rec[pos] = make_int2(((r & 15) << 12) | (w_col[e] & (IN_DIM - 1)),
                             __float_as_int(w_val[e]));
    }
}

// ---------------- pass 5: WMMA segment-GEMM + sigmoid ----------------
// One wave per (bucket, 16-row batch tile). Chunk of 4 edges:
//   A[m,k] = x[tileBase+m, col_k]            (16x4 f32)
//   B[k,n] = (rowlo_k == n) ? val_k : 0      (4x16 f32)
// Lane loads its two K-entries as one aligned b128 (2 fused records);
// B elements are built with bitwise masks (no conditional loads, EXEC stays 1s).
__global__ __launch_bounds__(32)
void NN_neat_wmma(const float* __restrict__ x,
                  const int* __restrict__ offsets,
                  const int4* __restrict__ rec4,
                  float* __restrict__ out) {
    const int bucket   = blockIdx.y;
    const int tileBase = blockIdx.x * 16;
    const int lane     = threadIdx.x;          // 0..31, wave32
    const int m        = lane & 15;            // M for A, N for B (ISA layout)
    const int hi       = (lane >> 4) & 1;      // lanes 16-31 hold K=2,3

    const unsigned rowOff = (unsigned)(tileBase + m) * IN_DIM;  // 32-bit: GVS scale_offset

    const int start = offsets[bucket];         // multiple of PADK
    const int end   = offsets[bucket + 1];

    v8f acc0 = {0.f, 0.f, 0.f, 0.f, 0.f, 0.f, 0.f, 0.f};
    v8f acc1 = acc0;

    const int eoff = 2 * hi;                   // upper half-wave takes K=2,3
    for (int base = start; base < end; base += 16) {
        __builtin_prefetch(&rec4[(unsigned)(base + 256) >> 1], 0, 1);  // edge stream
        #pragma unroll
        for (int c = 0; c < 4; ++c) {
            const unsigned q = (unsigned)(base + 4 * c + eoff) >> 1;   // int4 index
            const int4 r = rec4[q];            // one global_load_b128, GVS scale_offset
            // r.x/.z = packed{0,1}; r.y/.w = val bits{0,1}
            const int mask0 = -(int)((r.x >> 12) == m);   // all-ones iff this lane's col
            const int mask1 = -(int)((r.z >> 12) == m);
            v2f a, b;
            a.x = x[rowOff + (unsigned)(r.x & (IN_DIM - 1))];
            a.y = x[rowOff + (unsigned)(r.z & (IN_DIM - 1))];
            b.x = __int_as_float(r.y & mask0);            // exact 0.0f when masked
            b.y = __int_as_float(r.w & mask1);
            if (c & 1) {
                acc1 = __builtin_amdgcn_wmma_f32_16x16x4_f32(
                           false, a, false, b, (short)0, acc1, false, false);
            } else {
                acc0 = __builtin_amdgcn_wmma_f32_16x16x4_f32(
                           false, a, false, b, (short)0, acc0, false, false);
            }
        }
    }

    // C/D layout: VGPR i -> M = i + 8*hi, N = lane & 15
    const unsigned outCol = (unsigned)(bucket * 16 + m);
    const unsigned mBase  = (unsigned)(tileBase + (hi ? 8 : 0));
    #pragma unroll
    for (int i = 0; i < 8; ++i) {
        float v = acc0[i] + acc1[i];
        // sigmoid via fast v_exp_f32 + single v_rcp_f32 (no IEEE div sequence)
        out[(mBase + i) * OUT_DIM + outCol] =
            __builtin_amdgcn_rcpf(1.0f + __expf(-v));
    }
}

extern "C" void kernel_launch(void* const* d_in, const int* in_sizes, int n_in,
                              void* d_out, int out_size, void* d_ws, size_t ws_size,
                              hipStream_t stream) {
    const float* x        = (const float*)d_in[0];
    const float* w_values = (const float*)d_in[1];
    const int*   w_row    = (const int*)d_in[2];
    const int*   w_col    = (const int*)d_in[3];
    float*       out      = (float*)d_out;
    const int nnz = in_sizes[1];

    // workspace (ints): [0,256) counts | [256,513) offsets | [520,776) cursor
    //                   [1024, 1024 + 2*(nnz + PADK*NBUCKET)) fused records (int2)
    int*  counts  = (int*)d_ws;
    int*  offsets = counts + 256;
    int*  cursor  = counts + 520;
    int2* rec     = (int2*)(counts + 1024);

    NN_zero_counts  <<<1, 256, 0, stream>>>(counts);
    NN_count_edges  <<<(nnz + 255) / 256, 256, 0, stream>>>(w_row, counts, nnz);
    NN_scan_offsets <<<1, 256, 0, stream>>>(counts, offsets, cursor);
    NN_fill_pad     <<<1, 256, 0, stream>>>(counts, offsets, rec);
    NN_scatter_edges<<<(nnz + 255) / 256, 256, 0, stream>>>(w_row, w_col, w_values,
                                                            cursor, rec, nnz);
    dim3 grid(B_DIM / 16, NBUCKET);   // 16 batch tiles x 256 buckets, 1 wave each
    NN_neat_wmma<<<grid, 32, 0, stream>>>(x, offsets, (const int4*)rec, out);
}